// WAttnLayer_5583457485080
// MI455X (gfx1250) — compile-verified
//
#include <hip/hip_runtime.h>
#include <cstddef>

// Problem dims (fixed by the reference)
#define B_   4
#define N_   512
#define IN_  64
#define OUT_ 32
#define H_   4

typedef float v2f __attribute__((ext_vector_type(2)));
typedef float v4f __attribute__((ext_vector_type(4)));
typedef float v8f __attribute__((ext_vector_type(8)));

__device__ __forceinline__ float fast_tanh(float x) {
#if __has_builtin(__builtin_amdgcn_tanhf)
    return __builtin_amdgcn_tanhf(x);          // V_TANH_F32 (gfx1250 trans op)
#elif __has_builtin(__builtin_amdgcn_tanh_f32)
    return __builtin_amdgcn_tanh_f32(x);
#else
    return tanhf(x);
#endif
}

// ---------------------------------------------------------------------------
// Kernel 1: q/k/v projections with V_WMMA_F32_16X16X4_F32.
// grid = (row_tiles=128, h*2+coltile=8, which=3), block = 32 (one wave).
// ---------------------------------------------------------------------------
__global__ __launch_bounds__(32) void proj_kernel(
    const float* __restrict__ inp,
    const float* __restrict__ Wq, const float* __restrict__ Wk,
    const float* __restrict__ Wv,
    float* __restrict__ qws, float* __restrict__ kws, float* __restrict__ vws)
{
    const int rt    = blockIdx.x;        // 0..127  (16 rows of flattened B*N)
    const int hy    = blockIdx.y;        // 0..7    h*2 + coltile
    const int which = blockIdx.z;        // 0=q 1=k 2=v
    const int h  = hy >> 1;
    const int c0 = (hy & 1) * 16;
    const float* W  = (which == 0) ? Wq : (which == 1) ? Wk : Wv;
    float*      dst = (which == 0) ? qws : (which == 1) ? kws : vws;

    const int l  = threadIdx.x;
    const int m  = l & 15;               // M row within tile / N col within tile
    const int hi = l >> 4;               // lane half selects K pair
    const int r0 = rt * 16;
    const float* Wh = W + (size_t)h * IN_ * OUT_;

    v8f acc = {};
    #pragma unroll
    for (int k = 0; k < IN_; k += 4) {
        const int ka = k + hi * 2;
        v2f a, b;
        // A (16x4 f32): lanes 0-15 K={k,k+1}, lanes 16-31 K={k+2,k+3}
        a.x = inp[(size_t)(r0 + m) * IN_ + ka];
        a.y = inp[(size_t)(r0 + m) * IN_ + ka + 1];
        // B (4x16 f32): row K striped the same way, col = c0+m
        b.x = Wh[(size_t)ka * OUT_ + c0 + m];
        b.y = Wh[(size_t)(ka + 1) * OUT_ + c0 + m];
        acc = __builtin_amdgcn_wmma_f32_16x16x4_f32(
                  false, a, false, b, (short)0, acc, false, false);
    }

    float* dh = dst + (size_t)h * (B_ * N_ * OUT_);   // [2048, 32] per head
    #pragma unroll
    for (int g = 0; g < 8; ++g) {
        const int row = r0 + (hi ? 8 + g : g);        // D: VGPR g -> M=g / M=8+g
        dh[(size_t)row * OUT_ + c0 + m] = acc[g];
    }
}

// ---------------------------------------------------------------------------
// Kernel 2: fused score(tanh) -> softmax -> attn x val (WMMA) per
// (h, b, 16-row tile). 512 blocks x 128 threads (4 waves).
// ---------------------------------------------------------------------------
__global__ __launch_bounds__(128) void attn_kernel(
    const float* __restrict__ qws, const float* __restrict__ kws,
    const float* __restrict__ vws, const float* __restrict__ vvec,
    float* __restrict__ outp, float* __restrict__ attn,
    float* __restrict__ score)
{
    __shared__ __align__(16) float S[16][N_];     // 32 KB: score / attn block
    __shared__ __align__(16) float qt[16][OUT_];  // q tile
    __shared__ __align__(16) float vsh[OUT_];     // scoring vector
    __shared__ float red[16][8];                  // softmax partials
    __shared__ float rowmax[16];
    __shared__ float rowsum[16];
    __shared__ float Dp[4][32][8];                // WMMA partial accumulators

    const int bid = blockIdx.x;          // h*128 + b*32 + it
    const int it  = bid & 31;
    const int b   = (bid >> 5) & 3;
    const int h   = bid >> 7;
    const int t   = threadIdx.x;
    const int l   = t & 31;
    const int wid = t >> 5;
    const int i0  = it * 16;
    const int hb  = h * B_ + b;

    const float* qh = qws + ((size_t)h * B_ * N_ + (size_t)b * N_) * OUT_;
    const float* kh = kws + ((size_t)h * B_ * N_ + (size_t)b * N_) * OUT_;
    const float* vh = vws + ((size_t)h * B_ * N_ + (size_t)b * N_) * OUT_;

    // stage q tile + scoring vector in LDS
    for (int idx = t; idx < 16 * OUT_; idx += 128)
        qt[idx >> 5][idx & 31] = qh[(size_t)(i0 + (idx >> 5)) * OUT_ + (idx & 31)];
    if (t < OUT_) vsh[t] = vvec[h * OUT_ + t];
    __syncthreads();

    // ---- Phase A: scores s(i,j) = sum_o v[o]*tanh(q[i,o] + k[j,o]) ----
    {
        const int jm    = l & 15;
        const int ihalf = (l >> 4) * 8;

        // scoring vector -> registers (shared across all tiles)
        v4f vr[8];
        #pragma unroll
        for (int o4 = 0; o4 < 8; ++o4)
            vr[o4] = *(const v4f*)(&vsh[o4 * 4]);

        for (int tj = wid * 16; tj < N_; tj += 64) {
            const int j = tj + jm;
            // one k row -> registers, 8 x b128 loads
            v4f kr[8];
            #pragma unroll
            for (int o4 = 0; o4 < 8; ++o4)
                kr[o4] = *(const v4f*)(kh + (size_t)j * OUT_ + o4 * 4);
            if (tj + 64 < N_)   // global_prefetch next tile's k row
                __builtin_prefetch(kh + (size_t)(tj + 64 + jm) * OUT_, 0, 3);

            for (int p = 0; p < 8; ++p) {
                const int i = ihalf + p;
                float s = 0.f;
                #pragma unroll
                for (int o4 = 0; o4 < 8; ++o4) {
                    const v4f q4 = *(const v4f*)(&qt[i][o4 * 4]);
                    s += vr[o4].x * fast_tanh(q4.x + kr[o4].x);
                    s += vr[o4].y * fast_tanh(q4.y + kr[o4].y);
                    s += vr[o4].z * fast_tanh(q4.z + kr[o4].z);
                    s += vr[o4].w * fast_tanh(q4.w + kr[o4].w);
                }
                S[i][j] = s;
                score[((size_t)hb * N_ + i0 + i) * N_ + j] = s;
            }
        }
    }
    __syncthreads();

    // ---- Phase B: softmax over j (8 threads per row, float4 traffic) ----
    {
        const int row = t >> 3;
        const int seg = t & 7;
        const int cb  = seg * 64;

        float mx = -3.402823466e38f;
        for (int j4 = 0; j4 < 16; ++j4) {
            const v4f x = *(const v4f*)(&S[row][cb + j4 * 4]);
            mx = fmaxf(mx, fmaxf(fmaxf(x.x, x.y), fmaxf(x.z, x.w)));
        }
        red[row][seg] = mx;
        __syncthreads();
        if (seg == 0) {
            float m2 = red[row][0];
            for (int s2 = 1; s2 < 8; ++s2) m2 = fmaxf(m2, red[row][s2]);
            rowmax[row] = m2;
        }
        __syncthreads();
        const float rm = rowmax[row];
        float sum = 0.f;
        for (int j4 = 0; j4 < 16; ++j4) {
            v4f x = *(const v4f*)(&S[row][cb + j4 * 4]);
            x.x = __expf(x.x - rm); x.y = __expf(x.y - rm);
            x.z = __expf(x.z - rm); x.w = __expf(x.w - rm);
            *(v4f*)(&S[row][cb + j4 * 4]) = x;
            sum += (x.x + x.y) + (x.z + x.w);
        }
        red[row][seg] = sum;
        __syncthreads();
        if (seg == 0) {
            float s2 = 0.f;
            for (int k2 = 0; k2 < 8; ++k2) s2 += red[row][k2];
            rowsum[row] = s2;
        }
        __syncthreads();
        const float inv = 1.0f / rowsum[row];
        float* arow = attn + ((size_t)hb * N_ + i0 + row) * N_ + cb;
        for (int j4 = 0; j4 < 16; ++j4) {
            v4f x = *(const v4f*)(&S[row][cb + j4 * 4]);
            x.x *= inv; x.y *= inv; x.z *= inv; x.w *= inv;
            *(v4f*)(&S[row][cb + j4 * 4]) = x;
            *(v4f*)(arow + j4 * 4) = x;
        }
    }
    __syncthreads();

    // ---- Phase C: out = attn x val via WMMA f32 16x16x4 ----
    // wave -> (col tile c = wid&1, K half kh2 = wid>>1); every wave runs WMMA
    // with full EXEC, partials reduced through LDS.
    {
        const int c   = wid & 1;
        const int kh2 = wid >> 1;
        const int m   = l & 15;
        const int hi  = l >> 4;
        const int nn  = c * 16 + m;
        v8f acc = {};
        const int kbeg = kh2 * 256;
        for (int k = kbeg; k < kbeg + 256; k += 4) {
            const int ka = k + hi * 2;
            v2f a, bb;
            a.x  = S[m][ka];
            a.y  = S[m][ka + 1];
            bb.x = vh[(size_t)ka * OUT_ + nn];
            bb.y = vh[(size_t)(ka + 1) * OUT_ + nn];
            acc = __builtin_amdgcn_wmma_f32_16x16x4_f32(
                      false, a, false, bb, (short)0, acc, false, false);
        }
        #pragma unroll
        for (int g = 0; g < 8; ++g) Dp[wid][l][g] = acc[g];
        __syncthreads();
        if (wid < 2) {
            #pragma unroll
            for (int g = 0; g < 8; ++g) {
                const float v = Dp[wid][l][g] + Dp[wid + 2][l][g];
                const int rowi = (hi ? 8 + g : g);
                outp[(size_t)(b * N_ + i0 + rowi) * (H_ * OUT_)
                     + h * OUT_ + wid * 16 + m] = v;
            }
        }
    }
}

// ---------------------------------------------------------------------------
extern "C" void kernel_launch(void* const* d_in, const int* in_sizes, int n_in,
                              void* d_out, int out_size, void* d_ws, size_t ws_size,
                              hipStream_t stream) {
    (void)in_sizes; (void)n_in; (void)out_size; (void)ws_size;
    const float* inp  = (const float*)d_in[0];
    const float* Wq   = (const float*)d_in[1];
    const float* Wk   = (const float*)d_in[2];
    const float* Wv   = (const float*)d_in[3];
    const float* vvec = (const float*)d_in[4];

    float* out   = (float*)d_out;                               // [B,N,H*OUT]
    float* attn  = out  + (size_t)B_ * N_ * H_ * OUT_;          // [H,B,N,N]
    float* score = attn + (size_t)H_ * B_ * N_ * N_;            // [H,B,N,N]

    float* qws = (float*)d_ws;                                  // [H,B*N,OUT]
    float* kws = qws + (size_t)H_ * B_ * N_ * OUT_;
    float* vws = kws + (size_t)H_ * B_ * N_ * OUT_;

    dim3 g1((B_ * N_) / 16, H_ * 2, 3);
    proj_kernel<<<g1, 32, 0, stream>>>(inp, Wq, Wk, Wv, qws, kws, vws);

    attn_kernel<<<dim3(H_ * B_ * (N_ / 16)), 128, 0, stream>>>(
        qws, kws, vws, vvec, out, attn, score);
}